// LigerLinearCrossEntropyLoss_11046655885610
// MI455X (gfx1250) — compile-verified
//
#include <hip/hip_runtime.h>
#include <math.h>

// ---------------------------------------------------------------------------
// Fused linear + cross-entropy loss for MI455X (gfx1250, wave32, WMMA).
//   outputs [4,1024,2048] f32, targets [4,1024] i32, weight [32000,2048] f32
//   -> scalar mean loss (f32)
// Memory-bound op (262MB weight @ 23.3TB/s ~= 11us >> fp32-WMMA compute), so
// we keep full fp32 precision via V_WMMA_F32_16X16X4_F32 and optimize data
// movement: activations resident in LDS, one-pass online softmax, L2-shared
// weight streaming across token-tile workgroups.
// ---------------------------------------------------------------------------

typedef __attribute__((ext_vector_type(2))) float v2f;
typedef __attribute__((ext_vector_type(8))) float v8f;

#define NTOK        4096
#define DDIM        2048
#define VDIM        32000
#define TOK_TILE    32
#define VCHUNKS     5
#define VPER_CHUNK  (VDIM / VCHUNKS)      // 6400
#define VPER_ITER   256                   // 8 waves * 32 vocab cols
#define NITER       (VPER_CHUNK / VPER_ITER) // 25
#define AS_STRIDE   2052                  // 2048 + 4 pad -> conflict-free b64
#define IGNORE_IDX  (-100)

__global__ void lce_init(float* sumb, int* cntb) {
  if (threadIdx.x == 0) { sumb[0] = 0.0f; cntb[0] = 0; }
}

__launch_bounds__(256)
__global__ void lce_main(const float* __restrict__ x,
                         const int*   __restrict__ targets,
                         const float* __restrict__ w,
                         float* __restrict__ Pm,
                         float* __restrict__ Ps,
                         float* __restrict__ Tg)
{
  __shared__ __align__(16) float As[TOK_TILE * AS_STRIDE];   // ~257 KB
  __shared__ float SmM[8][TOK_TILE];
  __shared__ float SmS[8][TOK_TILE];

  const int tid  = threadIdx.x;
  const int lane = tid & 31;
  const int wave = tid >> 5;
  const int half = lane >> 4;       // wave32 half: C-tile M offset +8
  const int lr   = lane & 15;
  const int tile_base  = blockIdx.x * TOK_TILE;
  const int chunk      = blockIdx.y;
  const int chunk_base = chunk * VPER_CHUNK;

  // ---- stage activations: 32 tokens x 2048 f32, padded rows ----
  {
    const float4* xt = (const float4*)(x + (size_t)tile_base * DDIM);
    for (int i = tid; i < TOK_TILE * DDIM / 4; i += 256) {
      float4 v = xt[i];
      int e = i * 4;
      int row = e >> 11;            // /2048
      int col = e & (DDIM - 1);
      *(float4*)&As[row * AS_STRIDE + col] = v;
    }
  }
  __syncthreads();

  // per-lane targets for the 16 tokens this lane's half-wave owns
  int tg[2][8];
#pragma unroll
  for (int m = 0; m < 2; ++m)
#pragma unroll
    for (int r = 0; r < 8; ++r)
      tg[m][r] = targets[tile_base + m * 16 + r + 8 * half];

  // running online-softmax state (replicated across the 16 lanes of a half)
  float run_m[2][8], run_s[2][8];
#pragma unroll
  for (int m = 0; m < 2; ++m)
#pragma unroll
    for (int r = 0; r < 8; ++r) { run_m[m][r] = -INFINITY; run_s[m][r] = 0.0f; }

  // WMMA A operand: lane holds (M = lr, K = k + 2*half + {0,1}) -> b64 from LDS
  const float* a0p = &As[lr * AS_STRIDE + 2 * half];          // tokens 0..15
  const float* a1p = &As[(lr + 16) * AS_STRIDE + 2 * half];   // tokens 16..31

  for (int it = 0; it < NITER; ++it) {
    const int nb = chunk_base + it * VPER_ITER + wave * 32;   // wave's vocab base
    // WMMA B operand: lane holds (N = lr, K = k + 2*half + {0,1}) -> b64 global
    const float* b0p = w + (size_t)(nb + lr) * DDIM + 2 * half;
    const float* b1p = w + (size_t)(nb + 16 + lr) * DDIM + 2 * half;

    v8f z = {0,0,0,0,0,0,0,0};
    v8f acc00 = z, acc01 = z, acc10 = z, acc11 = z;

#pragma unroll 4
    for (int k = 0; k < DDIM; k += 4) {
      v2f a0 = *(const v2f*)(a0p + k);
      v2f a1 = *(const v2f*)(a1p + k);
      v2f b0 = *(const v2f*)(b0p + k);
      v2f b1 = *(const v2f*)(b1p + k);
      acc00 = __builtin_amdgcn_wmma_f32_16x16x4_f32(false, a0, false, b0, (short)0, acc00, false, false);
      acc01 = __builtin_amdgcn_wmma_f32_16x16x4_f32(false, a0, false, b1, (short)0, acc01, false, false);
      acc10 = __builtin_amdgcn_wmma_f32_16x16x4_f32(false, a1, false, b0, (short)0, acc10, false, false);
      acc11 = __builtin_amdgcn_wmma_f32_16x16x4_f32(false, a1, false, b1, (short)0, acc11, false, false);
    }

    // ---- online-softmax update over this 32-column block ----
#pragma unroll
    for (int m = 0; m < 2; ++m) {
#pragma unroll
      for (int r = 0; r < 8; ++r) {
        float v0 = (m == 0) ? acc00[r] : acc10[r];   // cols nb..nb+15
        float v1 = (m == 0) ? acc01[r] : acc11[r];   // cols nb+16..nb+31

        // block max across the 16 lanes of this half (width-16 butterfly)
        float bm = fmaxf(v0, v1);
#pragma unroll
        for (int off = 8; off >= 1; off >>= 1)
          bm = fmaxf(bm, __shfl_xor(bm, off, 16));

        float bs = __expf(v0 - bm) + __expf(v1 - bm);
#pragma unroll
        for (int off = 8; off >= 1; off >>= 1)
          bs += __shfl_xor(bs, off, 16);

        float om = run_m[m][r];
        float nm = fmaxf(om, bm);
        run_s[m][r] = run_s[m][r] * __expf(om - nm) + bs * __expf(bm - nm);
        run_m[m][r] = nm;

        // target logit: exactly one lane in the grid owns (token, target col)
        int c = tg[m][r] - nb;
        if (c >= 0 && c < 32) {
          float val = (c < 16) ? v0 : v1;
          if (lr == (c & 15))
            Tg[tile_base + m * 16 + r + 8 * half] = val;
        }
      }
    }
  }

  // ---- merge 8 wave partials per token via LDS ----
#pragma unroll
  for (int m = 0; m < 2; ++m)
#pragma unroll
    for (int r = 0; r < 8; ++r)
      if (lr == r) {
        int tl = m * 16 + r + 8 * half;
        SmM[wave][tl] = run_m[m][r];
        SmS[wave][tl] = run_s[m][r];
      }
  __syncthreads();

  if (tid < TOK_TILE) {
    float M = -INFINITY, S = 0.0f;
#pragma unroll
    for (int wv = 0; wv < 8; ++wv) {
      float mi = SmM[wv][tid], si = SmS[wv][tid];
      float M2 = fmaxf(M, mi);
      S = S * __expf(M - M2) + si * __expf(mi - M2);
      M = M2;
    }
    Pm[chunk * NTOK + tile_base + tid] = M;
    Ps[chunk * NTOK + tile_base + tid] = S;
  }
}

__global__ void lce_finalize(const float* __restrict__ Pm,
                             const float* __restrict__ Ps,
                             const float* __restrict__ Tg,
                             const int*   __restrict__ targets,
                             float* sumb, int* cntb)
{
  int i = blockIdx.x * blockDim.x + threadIdx.x;
  if (i >= NTOK) return;
  float M = -INFINITY, S = 0.0f;
#pragma unroll
  for (int c = 0; c < VCHUNKS; ++c) {
    float mi = Pm[c * NTOK + i], si = Ps[c * NTOK + i];
    float M2 = fmaxf(M, mi);
    S = S * __expf(M - M2) + si * __expf(mi - M2);
    M = M2;
  }
  int t = targets[i];
  if (t != IGNORE_IDX) {
    float lse = M + __logf(S);
    atomicAdd(sumb, lse - Tg[i]);
    atomicAdd(cntb, 1);
  }
}

__global__ void lce_div(const float* sumb, const int* cntb, float* out) {
  int c = cntb[0];
  out[0] = sumb[0] / (float)(c > 0 ? c : 1);
}

extern "C" void kernel_launch(void* const* d_in, const int* in_sizes, int n_in,
                              void* d_out, int out_size, void* d_ws, size_t ws_size,
                              hipStream_t stream) {
  const float* x  = (const float*)d_in[0];
  const int*   t  = (const int*)  d_in[1];
  const float* w  = (const float*)d_in[2];
  float* out = (float*)d_out;

  float* Pm   = (float*)d_ws;                 // [VCHUNKS][NTOK]
  float* Ps   = Pm + (size_t)VCHUNKS * NTOK;  // [VCHUNKS][NTOK]
  float* Tg   = Ps + (size_t)VCHUNKS * NTOK;  // [NTOK]
  float* sumb = Tg + NTOK;                    // [1]
  int*   cntb = (int*)(sumb + 1);             // [1]

  lce_init<<<1, 1, 0, stream>>>(sumb, cntb);
  lce_main<<<dim3(NTOK / TOK_TILE, VCHUNKS), 256, 0, stream>>>(x, t, w, Pm, Ps, Tg);
  lce_finalize<<<(NTOK + 255) / 256, 256, 0, stream>>>(Pm, Ps, Tg, t, sumb, cntb);
  lce_div<<<1, 1, 0, stream>>>(sumb, cntb, out);
}